// SourceAttributionGNN_37366215475273
// MI455X (gfx1250) — compile-verified
//
#include <hip/hip_runtime.h>
#include <hip/hip_bf16.h>
#include <math.h>

// ---------------------------------------------------------------------------
// 2-layer GAT on MI455X (gfx1250, wave32).
//  - GEMM1 (K=4) via V_WMMA_F32_16X16X4_F32 (exact fp32, perfect K fit).
//  - Attention softmax via encoded-float atomicMax + atomicAdd (L2 atomics;
//    all node arrays fit in the 192MB L2, so edge passes are L2-bound).
//  - b128 vector loads on the bandwidth-carrying gather paths.
// ---------------------------------------------------------------------------

typedef __attribute__((ext_vector_type(2))) float v2f;
typedef __attribute__((ext_vector_type(8))) float v8f;

#define NEG_SLOPE 0.2f
#define EPSV 1e-16f

__device__ __forceinline__ float lrelu(float x) { return x > 0.f ? x : NEG_SLOPE * x; }

// Monotone float -> uint mapping so unsigned atomicMax == float max.
__device__ __forceinline__ unsigned fenc(float x) {
    unsigned u = __float_as_uint(x);
    return (u & 0x80000000u) ? ~u : (u | 0x80000000u);
}
__device__ __forceinline__ float fdec(unsigned u) {
    return (u & 0x80000000u) ? __uint_as_float(u & 0x7FFFFFFFu) : __uint_as_float(~u);
}

// ---------------------------------------------------------------------------
// K0: zero the accumulator / max / sum region (float4-wide).
// ---------------------------------------------------------------------------
__global__ void k_zero(float4* __restrict__ p, int n4) {
    int i = blockIdx.x * blockDim.x + threadIdx.x;
    if (i < n4) p[i] = make_float4(0.f, 0.f, 0.f, 0.f);
}

// ---------------------------------------------------------------------------
// K1: h1 = x @ W1   ([N,4] @ [4,64]) with WMMA f32 16x16x4.
// One wave per 16-row tile; 4 WMMAs (one per 16-col tile of the 64 outputs).
// A layout: lanes 0..15 hold M=0..15; VGPR0/1 = K=(0,1) for low half,
// (2,3) for high half. C/D: VGPR v -> row v (lanes<16) / v+8 (lanes>=16).
// Full tiles take a branch-free store path (wave-uniform decision) so EXEC
// stays all-1s and no per-lane exec save/restore surrounds the WMMAs.
// ---------------------------------------------------------------------------
__global__ void k_gemm1_wmma(const float* __restrict__ x,
                             const float* __restrict__ W1,
                             float* __restrict__ h1, int N, int NT) {
    int wave = (blockIdx.x * blockDim.x + threadIdx.x) >> 5;
    int lane = threadIdx.x & 31;
    if (wave >= NT) return;  // uniform per wave

    int row   = wave * 16;
    int mrow  = lane & 15;
    int khalf = (lane >> 4) << 1;  // 0 or 2

    int r = row + mrow;
    if (r > N - 1) r = N - 1;  // branchless clamp for tail tiles
    v2f a;
    a.x = x[(size_t)r * 4 + khalf + 0];
    a.y = x[(size_t)r * 4 + khalf + 1];

    int  rbase = row + ((lane >> 4) << 3);  // D VGPR v -> row rbase+v
    bool full  = (row + 16 <= N);           // wave-uniform

#pragma unroll
    for (int nt = 0; nt < 4; ++nt) {
        int n = nt * 16 + mrow;
        v2f b;
        b.x = W1[(khalf + 0) * 64 + n];
        b.y = W1[(khalf + 1) * 64 + n];
        v8f c = {};
        c = __builtin_amdgcn_wmma_f32_16x16x4_f32(
                /*neg_a=*/false, a, /*neg_b=*/false, b,
                /*c_mod=*/(short)0, c, /*reuse_a=*/false, /*reuse_b=*/false);
        if (full) {
            float* hp = h1 + (size_t)rbase * 64 + n;
#pragma unroll
            for (int v = 0; v < 8; ++v) hp[(size_t)v * 64] = c[v];
        } else {
#pragma unroll
            for (int v = 0; v < 8; ++v) {
                int rr = rbase + v;
                if (rr < N) h1[(size_t)rr * 64 + n] = c[v];
            }
        }
    }
}

// ---------------------------------------------------------------------------
// K2: per-node attention logits for layer 1: alpha_s/d[n][h] = h1[n,h,:]·a
// ---------------------------------------------------------------------------
__global__ void k_alpha1(const float* __restrict__ h1,
                         const float* __restrict__ a_s,  // [4,16]
                         const float* __restrict__ a_d,  // [4,16]
                         float* __restrict__ as1, float* __restrict__ ad1, int N) {
    int n = blockIdx.x * blockDim.x + threadIdx.x;
    if (n >= N) return;
    const float4* hp = (const float4*)(h1 + (size_t)n * 64);
    const float4* asw = (const float4*)a_s;
    const float4* adw = (const float4*)a_d;
#pragma unroll
    for (int h = 0; h < 4; ++h) {
        float s = 0.f, d = 0.f;
#pragma unroll
        for (int q = 0; q < 4; ++q) {
            float4 v = hp[h * 4 + q];
            float4 ws = asw[h * 4 + q];
            float4 wd = adw[h * 4 + q];
            s += v.x * ws.x + v.y * ws.y + v.z * ws.z + v.w * ws.w;
            d += v.x * wd.x + v.y * wd.y + v.z * wd.z + v.w * wd.w;
        }
        as1[n * 4 + h] = s;
        ad1[n * 4 + h] = d;
    }
}

// Edge helpers: edges 0..E-1 are real, E..E+N-1 are self loops.
__device__ __forceinline__ void edge_sd(int e, int E,
                                        const int* __restrict__ src,
                                        const int* __restrict__ dst,
                                        int& s, int& d) {
    if (e < E) { s = src[e]; d = dst[e]; }
    else       { s = e - E;  d = e - E;  }
}

// ---------------------------------------------------------------------------
// K3/K4/K5: layer-1 edge passes (thread per (edge, head)).
// ---------------------------------------------------------------------------
__global__ void k_edge_max1(const int* __restrict__ src, const int* __restrict__ dst,
                            const float* __restrict__ as1, const float* __restrict__ ad1,
                            unsigned* __restrict__ m1, int E, int ETOT) {
    int t = blockIdx.x * blockDim.x + threadIdx.x;
    if (t >= ETOT * 4) return;
    int e = t >> 2, h = t & 3, s, d;
    edge_sd(e, E, src, dst, s, d);
    float v = lrelu(as1[s * 4 + h] + ad1[d * 4 + h]);
    atomicMax(&m1[d * 4 + h], fenc(v));
}

__global__ void k_edge_sum1(const int* __restrict__ src, const int* __restrict__ dst,
                            const float* __restrict__ as1, const float* __restrict__ ad1,
                            const unsigned* __restrict__ m1, float* __restrict__ s1,
                            int E, int ETOT) {
    int t = blockIdx.x * blockDim.x + threadIdx.x;
    if (t >= ETOT * 4) return;
    int e = t >> 2, h = t & 3, s, d;
    edge_sd(e, E, src, dst, s, d);
    float v = lrelu(as1[s * 4 + h] + ad1[d * 4 + h]);
    atomicAdd(&s1[d * 4 + h], expf(v - fdec(m1[d * 4 + h])));
}

__global__ void k_edge_agg1(const int* __restrict__ src, const int* __restrict__ dst,
                            const float* __restrict__ as1, const float* __restrict__ ad1,
                            const unsigned* __restrict__ m1, const float* __restrict__ s1,
                            const float* __restrict__ h1, float* __restrict__ out1,
                            int E, int ETOT) {
    int t = blockIdx.x * blockDim.x + threadIdx.x;
    if (t >= ETOT * 4) return;
    int e = t >> 2, h = t & 3, s, d;
    edge_sd(e, E, src, dst, s, d);
    float v     = lrelu(as1[s * 4 + h] + ad1[d * 4 + h]);
    float alpha = expf(v - fdec(m1[d * 4 + h])) / (s1[d * 4 + h] + EPSV);
    const float4* hs = (const float4*)(h1 + (size_t)s * 64 + h * 16);  // 16B aligned
    float*        od = out1 + (size_t)d * 64 + h * 16;
#pragma unroll
    for (int q = 0; q < 4; ++q) {
        float4 hv = hs[q];  // global_load_b128
        atomicAdd(&od[q * 4 + 0], alpha * hv.x);
        atomicAdd(&od[q * 4 + 1], alpha * hv.y);
        atomicAdd(&od[q * 4 + 2], alpha * hv.z);
        atomicAdd(&od[q * 4 + 3], alpha * hv.w);
    }
}

// ---------------------------------------------------------------------------
// K6: layer-1 finalize (bias + ELU) fused with GEMM2 (K=64, N=2, scalar)
// and layer-2 attention logits.
// ---------------------------------------------------------------------------
__global__ void k_fin1(const float* __restrict__ out1, const float* __restrict__ b1,
                       const float* __restrict__ W2,                 // [64,2]
                       const float* __restrict__ as2w,               // [1,2]
                       const float* __restrict__ ad2w,               // [1,2]
                       float* __restrict__ h2, float* __restrict__ as2,
                       float* __restrict__ ad2, int N) {
    int n = blockIdx.x * blockDim.x + threadIdx.x;
    if (n >= N) return;
    const float4* op = (const float4*)(out1 + (size_t)n * 64);
    const float4* bp = (const float4*)b1;
    float t0 = 0.f, t1 = 0.f;
#pragma unroll
    for (int q = 0; q < 16; ++q) {
        float4 ov = op[q];
        float4 bv = bp[q];
        float vv[4] = {ov.x + bv.x, ov.y + bv.y, ov.z + bv.z, ov.w + bv.w};
#pragma unroll
        for (int j = 0; j < 4; ++j) {
            float v = vv[j];
            v = v > 0.f ? v : (expf(v) - 1.0f);  // ELU
            int c = q * 4 + j;
            t0 += v * W2[c * 2 + 0];
            t1 += v * W2[c * 2 + 1];
        }
    }
    h2[n * 2 + 0] = t0;
    h2[n * 2 + 1] = t1;
    as2[n] = t0 * as2w[0] + t1 * as2w[1];
    ad2[n] = t0 * ad2w[0] + t1 * ad2w[1];
}

// ---------------------------------------------------------------------------
// K7/K8/K9: layer-2 edge passes (thread per edge, 1 head).
// ---------------------------------------------------------------------------
__global__ void k_edge_max2(const int* __restrict__ src, const int* __restrict__ dst,
                            const float* __restrict__ as2, const float* __restrict__ ad2,
                            unsigned* __restrict__ m2, int E, int ETOT) {
    int e = blockIdx.x * blockDim.x + threadIdx.x;
    if (e >= ETOT) return;
    int s, d;
    edge_sd(e, E, src, dst, s, d);
    atomicMax(&m2[d], fenc(lrelu(as2[s] + ad2[d])));
}

__global__ void k_edge_sum2(const int* __restrict__ src, const int* __restrict__ dst,
                            const float* __restrict__ as2, const float* __restrict__ ad2,
                            const unsigned* __restrict__ m2, float* __restrict__ s2,
                            int E, int ETOT) {
    int e = blockIdx.x * blockDim.x + threadIdx.x;
    if (e >= ETOT) return;
    int s, d;
    edge_sd(e, E, src, dst, s, d);
    float v = lrelu(as2[s] + ad2[d]);
    atomicAdd(&s2[d], expf(v - fdec(m2[d])));
}

__global__ void k_edge_agg2(const int* __restrict__ src, const int* __restrict__ dst,
                            const float* __restrict__ as2, const float* __restrict__ ad2,
                            const unsigned* __restrict__ m2, const float* __restrict__ s2,
                            const float* __restrict__ h2, float* __restrict__ out2,
                            int E, int ETOT) {
    int e = blockIdx.x * blockDim.x + threadIdx.x;
    if (e >= ETOT) return;
    int s, d;
    edge_sd(e, E, src, dst, s, d);
    float v     = lrelu(as2[s] + ad2[d]);
    float alpha = expf(v - fdec(m2[d])) / (s2[d] + EPSV);
    atomicAdd(&out2[d * 2 + 0], alpha * h2[s * 2 + 0]);
    atomicAdd(&out2[d * 2 + 1], alpha * h2[s * 2 + 1]);
}

// ---------------------------------------------------------------------------
// K10: bias + log_softmax over the 2 classes.
// ---------------------------------------------------------------------------
__global__ void k_final(const float* __restrict__ out2, const float* __restrict__ b2,
                        float* __restrict__ out, int N) {
    int n = blockIdx.x * blockDim.x + threadIdx.x;
    if (n >= N) return;
    float o0 = out2[n * 2 + 0] + b2[0];
    float o1 = out2[n * 2 + 1] + b2[1];
    float mx = fmaxf(o0, o1);
    float l  = logf(expf(o0 - mx) + expf(o1 - mx));
    out[n * 2 + 0] = o0 - mx - l;
    out[n * 2 + 1] = o1 - mx - l;
}

// ---------------------------------------------------------------------------
extern "C" void kernel_launch(void* const* d_in, const int* in_sizes, int n_in,
                              void* d_out, int out_size, void* d_ws, size_t ws_size,
                              hipStream_t stream) {
    const float* x    = (const float*)d_in[0];
    const int*   ei   = (const int*)d_in[1];
    const float* W1   = (const float*)d_in[2];
    const float* a_s1 = (const float*)d_in[3];
    const float* a_d1 = (const float*)d_in[4];
    const float* b1   = (const float*)d_in[5];
    const float* W2   = (const float*)d_in[6];
    const float* a_s2 = (const float*)d_in[7];
    const float* a_d2 = (const float*)d_in[8];
    const float* b2   = (const float*)d_in[9];

    const int N    = in_sizes[0] / 4;   // 100000
    const int E    = in_sizes[1] / 2;   // 3200000
    const int ETOT = E + N;             // + self loops
    const int* src = ei;
    const int* dst = ei + E;

    // Workspace layout (floats). Zero-required region is contiguous.
    float*    ws   = (float*)d_ws;
    float*    h1   = ws;                                   // N*64
    float*    out1 = h1 + (size_t)N * 64;                  // N*64  (zeroed)
    float*    s1   = out1 + (size_t)N * 64;                // N*4   (zeroed)
    unsigned* m1   = (unsigned*)(s1 + (size_t)N * 4);      // N*4   (zeroed)
    float*    s2   = (float*)(m1 + (size_t)N * 4);         // N     (zeroed)
    unsigned* m2   = (unsigned*)(s2 + N);                  // N     (zeroed)
    float*    out2 = (float*)(m2 + N);                     // N*2   (zeroed)
    float*    as1  = out2 + (size_t)N * 2;                 // N*4
    float*    ad1  = as1 + (size_t)N * 4;                  // N*4
    float*    h2   = ad1 + (size_t)N * 4;                  // N*2
    float*    as2  = h2 + (size_t)N * 2;                   // N
    float*    ad2  = as2 + N;                              // N

    const int TB = 256;
    // Zero region: out1..out2 = N*(64+4+4+1+1+2) = N*76 floats.
    {
        int nzero = N * 76;
        int n4 = nzero / 4;
        k_zero<<<(n4 + TB - 1) / TB, TB, 0, stream>>>((float4*)out1, n4);
    }
    // GEMM1 via WMMA: one wave per 16-row tile.
    {
        int NT = (N + 15) / 16;
        int threads = NT * 32;
        k_gemm1_wmma<<<(threads + TB - 1) / TB, TB, 0, stream>>>(x, W1, h1, N, NT);
    }
    k_alpha1<<<(N + TB - 1) / TB, TB, 0, stream>>>(h1, a_s1, a_d1, as1, ad1, N);

    int e4 = ETOT * 4;
    k_edge_max1<<<(e4 + TB - 1) / TB, TB, 0, stream>>>(src, dst, as1, ad1, m1, E, ETOT);
    k_edge_sum1<<<(e4 + TB - 1) / TB, TB, 0, stream>>>(src, dst, as1, ad1, m1, s1, E, ETOT);
    k_edge_agg1<<<(e4 + TB - 1) / TB, TB, 0, stream>>>(src, dst, as1, ad1, m1, s1, h1, out1, E, ETOT);

    k_fin1<<<(N + TB - 1) / TB, TB, 0, stream>>>(out1, b1, W2, a_s2, a_d2, h2, as2, ad2, N);

    k_edge_max2<<<(ETOT + TB - 1) / TB, TB, 0, stream>>>(src, dst, as2, ad2, m2, E, ETOT);
    k_edge_sum2<<<(ETOT + TB - 1) / TB, TB, 0, stream>>>(src, dst, as2, ad2, m2, s2, E, ETOT);
    k_edge_agg2<<<(ETOT + TB - 1) / TB, TB, 0, stream>>>(src, dst, as2, ad2, m2, s2, h2, out2, E, ETOT);

    k_final<<<(N + TB - 1) / TB, TB, 0, stream>>>(out2, b2, (float*)d_out, N);
}